// TangentGCN_81338090651948
// MI455X (gfx1250) — compile-verified
//
#include <hip/hip_runtime.h>

#define NUM_NODES 50000
#define DIM       128
#define NUM_EDGES 800000

typedef __attribute__((ext_vector_type(2))) float v2f;
typedef __attribute__((ext_vector_type(8))) float v8f;

// ---------------------------------------------------------------------------
// Zero the aggregation buffer (float4 stores, fully coalesced).
// ---------------------------------------------------------------------------
__global__ __launch_bounds__(256) void zero_kernel(float* __restrict__ p, int n4) {
    int i = blockIdx.x * blockDim.x + threadIdx.x;
    if (i < n4) {
        ((float4*)p)[i] = make_float4(0.f, 0.f, 0.f, 0.f);
    }
}

// ---------------------------------------------------------------------------
// SpMM: agg[dst] += val * h[src].  One wave32 per edge; each lane owns a
// contiguous float4 chunk of the 128-dim row (32 lanes * 4 = 128).
// Gathers are B128 coalesced; scatter uses global_atomic_add_f32 without
// return (STOREcnt path).  All data is L2-resident (h = 25.6 MB << 192 MB).
// ---------------------------------------------------------------------------
__global__ __launch_bounds__(256) void spmm_kernel(const int*   __restrict__ esrc,
                                                   const int*   __restrict__ edst,
                                                   const float* __restrict__ eval,
                                                   const float* __restrict__ h,
                                                   float*       __restrict__ agg) {
    int t    = blockIdx.x * blockDim.x + threadIdx.x;
    int e    = t >> 5;        // edge index (one wave per edge)
    int lane = t & 31;        // float4 chunk within the 128-dim row
    if (e >= NUM_EDGES) return;

    int   s = esrc[e];
    int   d = edst[e];
    float v = eval[e];

    const float4 m = ((const float4*)(h + (size_t)s * DIM))[lane];
    float* o = agg + (size_t)d * DIM + lane * 4;
    atomicAdd(o + 0, m.x * v);
    atomicAdd(o + 1, m.y * v);
    atomicAdd(o + 2, m.z * v);
    atomicAdd(o + 3, m.w * v);
}

// ---------------------------------------------------------------------------
// out = relu(A @ W^T + b), A:[N,128] f32, W:[128,128] f32.
// One wave computes one 16x16 f32 tile via V_WMMA_F32_16X16X4_F32 (K-loop of
// 32 steps).  8 waves per block cover the 8 column tiles of one 16-row stripe.
//
// Fragment maps (ISA 7.12.2, 32-bit):
//   A 16x4 : lane<16 -> M=lane,    K base 0 ; lane>=16 -> M=lane-16, K base 2
//            VGPR0 = K+0, VGPR1 = K+1
//   B 4x16 : N = lane&15, K base = (lane>=16)?2:0 ; B[k][n] = W[n][k]
//   C 16x16: VGPR r -> M = r + 8*(lane>=16), N = lane&15
// ---------------------------------------------------------------------------
__global__ __launch_bounds__(256) void gemm_bias_relu_kernel(const float* __restrict__ A,
                                                             const float* __restrict__ W,
                                                             const float* __restrict__ bias,
                                                             float*       __restrict__ out) {
    const int lane = threadIdx.x & 31;
    const int wave = threadIdx.x >> 5;          // 0..7 -> column tile
    const int row0 = blockIdx.x * 16;
    const int col0 = wave * 16;

    const int half = lane >> 4;                 // 0 or 1
    const int idx  = lane & 15;                 // M (for A) / N (for B,C)
    const int koff = half * 2;                  // K base within the 4-chunk

    // C init: every acc VGPR holds value for N = col0+idx -> broadcast bias.
    const float bv = bias[col0 + idx];
    v8f acc = { bv, bv, bv, bv, bv, bv, bv, bv };

    const float* arow = A + (size_t)(row0 + idx) * DIM;   // A[M=idx][*]
    const float* wrow = W + (size_t)(col0 + idx) * DIM;   // W[N=idx][*] = B[*][N]

    #pragma unroll 4
    for (int k0 = 0; k0 < DIM; k0 += 4) {
        v2f af, bf;
        af.x = arow[k0 + koff];
        af.y = arow[k0 + koff + 1];
        bf.x = wrow[k0 + koff];
        bf.y = wrow[k0 + koff + 1];
        acc = __builtin_amdgcn_wmma_f32_16x16x4_f32(
            /*neg_a=*/false, af, /*neg_b=*/false, bf,
            /*c_mod=*/(short)0, acc, /*reuse_a=*/false, /*reuse_b=*/false);
    }

    // ReLU + store per C layout.
    #pragma unroll
    for (int r = 0; r < 8; ++r) {
        float x = acc[r];
        x = x > 0.f ? x : 0.f;
        out[(size_t)(row0 + r + half * 8) * DIM + col0 + idx] = x;
    }
}

// ---------------------------------------------------------------------------
// Pipeline: (zero -> spmm -> gemm) x 2 layers.  Workspace: agg + h1 buffers
// (2 * 50000 * 128 * 4 B = 51.2 MB).
// ---------------------------------------------------------------------------
extern "C" void kernel_launch(void* const* d_in, const int* in_sizes, int n_in,
                              void* d_out, int out_size, void* d_ws, size_t ws_size,
                              hipStream_t stream) {
    const int*   edge_src = (const int*)  d_in[0];
    const int*   edge_dst = (const int*)  d_in[1];
    const float* edge_val = (const float*)d_in[2];
    const float* emb      = (const float*)d_in[3];
    const float* W1       = (const float*)d_in[4];
    const float* b1       = (const float*)d_in[5];
    const float* W2       = (const float*)d_in[6];
    const float* b2       = (const float*)d_in[7];
    float*       out      = (float*)d_out;

    float* agg = (float*)d_ws;
    float* h1  = agg + (size_t)NUM_NODES * DIM;

    const int n4          = NUM_NODES * DIM / 4;          // 1.6M float4
    const int zero_blocks = (n4 + 255) / 256;
    const int spmm_blocks = (NUM_EDGES * 32 + 255) / 256; // 1 wave / edge
    const int gemm_blocks = NUM_NODES / 16;               // 3125 (exact)

    // Layer 1
    zero_kernel<<<zero_blocks, 256, 0, stream>>>(agg, n4);
    spmm_kernel<<<spmm_blocks, 256, 0, stream>>>(edge_src, edge_dst, edge_val, emb, agg);
    gemm_bias_relu_kernel<<<gemm_blocks, 256, 0, stream>>>(agg, W1, b1, h1);

    // Layer 2
    zero_kernel<<<zero_blocks, 256, 0, stream>>>(agg, n4);
    spmm_kernel<<<spmm_blocks, 256, 0, stream>>>(edge_src, edge_dst, edge_val, h1, agg);
    gemm_bias_relu_kernel<<<gemm_blocks, 256, 0, stream>>>(agg, W2, b2, out);
}